// Goggle_90744069030337
// MI455X (gfx1250) — compile-verified
//
#include <hip/hip_runtime.h>
#include <math.h>

// ---------------------------------------------------------------------------
// GOGGLE decoder forward, CDNA5 (gfx1250), fp32 WMMA 16x16x4.
//   B=1024 batch, N=256 nodes, D=64.
// Key simplification: concat(z, eye) @ emb_W  ==  z[n]*emb_W[n,0,:] + emb_W[n,1+n,:]
// All WMMA A/B fragments are single aligned 8-byte loads (K-pair-interleaved
// operand layouts) so the inner loop is load_b64 + v_wmma with no repacking.
// ---------------------------------------------------------------------------

typedef __attribute__((ext_vector_type(2))) float v2f;
typedef __attribute__((ext_vector_type(8))) float v8f;

#define GN 256   // nodes
#define GD 64    // feature dim

// workspace layout (float offsets)
#define WS_ADJ   0         // 256*256        (for degree norms)
#define WS_ADJP  65536     // 256*256        K-pair-packed adj^T operand
#define WS_NSRC  131072    // 256
#define WS_NDST  131328    // 256
#define WS_W0    131584    // 256*64
#define WS_CMAT  147968    // 256*64

// LDS pair-row strides (chosen == 32 mod 64 banks: half-waves hit disjoint banks)
#define BZ_STRIDE 160      // 128 pair-rows x 160  (useful 128)  -> 80 KB
#define H1_STRIDE 96       // 128 pair-rows x 96   (useful 64)   -> within bz region
#define AG_STRIDE 66       // agg1 row stride (even -> aligned b64 A-frags)

// ---------------------------------------------------------------------------
// Pre-pass kernels (run once per launch; outputs live in L2 for main kernel)
// ---------------------------------------------------------------------------
__global__ void gg_build_adj(const float* __restrict__ gl,
                             const int* __restrict__ it_num,
                             float* __restrict__ adj,
                             float* __restrict__ adjp) {
    int idx = blockIdx.x * blockDim.x + threadIdx.x;   // 65536 threads
    int i = idx >> 8, j = idx & 255;                   // adj[i][j]
    float v = (i == j) ? 1.0f : 1.0f / (1.0f + expf(-gl[idx]));
    if (it_num[0] > 50 && !(v > 0.1f)) v = 0.0f;       // nn.Threshold(0.1, 0)
    adj[idx] = v;
    // K-pair-packed layout for WMMA A fragments: pair over row index i
    adjp[(size_t)((i >> 1) * GN + j) * 2 + (i & 1)] = v;
}

__global__ void gg_build_norms(const float* __restrict__ adj,
                               float* __restrict__ nsrc,
                               float* __restrict__ ndst) {
    int t = threadIdx.x;                               // 256 threads
    float co = 0.0f, ci = 0.0f;
    for (int k = 0; k < GN; ++k) {
        co += (adj[t * GN + k] != 0.0f) ? 1.0f : 0.0f; // out-degree (row)
        ci += (adj[k * GN + t] != 0.0f) ? 1.0f : 0.0f; // in-degree (col)
    }
    nsrc[t] = rsqrtf(co);
    ndst[t] = rsqrtf(ci);
}

__global__ void gg_build_emb(const float* __restrict__ emb_W,
                             const float* __restrict__ emb_b,
                             float* __restrict__ W0,
                             float* __restrict__ Cmat) {
    int idx = blockIdx.x * blockDim.x + threadIdx.x;   // 16384 threads
    int n = idx >> 6, d = idx & 63;
    const float* base = emb_W + (size_t)n * 257 * GD;  // emb_W[n, f, d]
    W0[idx]   = base[d];                               // f = 0 (z channel)
    Cmat[idx] = base[(size_t)(1 + n) * GD + d] + emb_b[idx];  // one-hot f = 1+n
}

// ---------------------------------------------------------------------------
// Main fused kernel: one workgroup (16 wave32) per batch element.
// ---------------------------------------------------------------------------
__global__ void __launch_bounds__(512)
gg_main(const float* __restrict__ x,     const float* __restrict__ eps,
        const float* __restrict__ W_enc, const float* __restrict__ b_enc,
        const float* __restrict__ W_mu,  const float* __restrict__ b_mu,
        const float* __restrict__ W_lv,  const float* __restrict__ b_lv,
        const float* __restrict__ gc1_W, const float* __restrict__ gc1_b,
        const float* __restrict__ gc2_W, const float* __restrict__ gc2_b,
        const float* __restrict__ adjp,  const float* __restrict__ nsrc,
        const float* __restrict__ ndst,  const float* __restrict__ W0,
        const float* __restrict__ Cmat,  float* __restrict__ out) {
    __shared__ float s_x[GN];
    __shared__ float s_h[GD];
    __shared__ float s_z[GN];
    __shared__ float s_bz[128 * BZ_STRIDE];   // K-pair-packed bz*nsrc; reused as h1 pairs
    __shared__ float s_buf[GN * AG_STRIDE];   // agg1 [m][k]; reused as agg2 (stride 33)
    __shared__ float s_w1p[2048];             // gc1_W K-pair-packed (64x32)

    const int b    = blockIdx.x;
    const int t    = threadIdx.x;
    const int wave = t >> 5;
    const int lane = t & 31;
    const int l16  = lane & 15;
    const int hi   = lane >> 4;           // 0: lanes 0-15, 1: lanes 16-31
    const int kh   = hi * 2;              // K sub-offset for A/B fragments (f32 16x16x4)

    // ---- encoder: h = relu(x @ W_enc + b_enc) --------------------------------
    if (t < GN) s_x[t] = x[b * GN + t];
    __syncthreads();

    if (t < GD) {
        float acc = b_enc[t];
        for (int n = 0; n < GN; ++n) acc = fmaf(s_x[n], W_enc[n * GD + t], acc);
        s_h[t] = fmaxf(acc, 0.0f);
    }
    __syncthreads();

    // ---- mu/logvar, reparameterize ------------------------------------------
    if (t < GN) {
        float mu = b_mu[t], lv = b_lv[t];
        for (int d = 0; d < GD; ++d) {
            float hv = s_h[d];
            mu = fmaf(hv, W_mu[d * GN + t], mu);
            lv = fmaf(hv, W_lv[d * GN + t], lv);
        }
        s_z[t] = mu + eps[b * GN + t] * expf(0.5f * lv);
    }
    __syncthreads();

    // ---- collapsed one-hot embedding -> pair-packed LDS, norm_src folded -----
    for (int i = t; i < GN * GD; i += 512) {
        int n = i >> 6, d = i & 63;
        float v = tanhf(fmaf(s_z[n], W0[i], Cmat[i])) * nsrc[n];
        s_bz[(n >> 1) * BZ_STRIDE + d * 2 + (n & 1)] = v;
    }
    // stage gc1_W pair-packed (consumed by GEMM2)
    for (int i = t; i < GD * 32; i += 512) {
        int k = i >> 5, j = i & 31;
        s_w1p[(k >> 1) * 64 + j * 2 + (k & 1)] = gc1_W[i];
    }
    __syncthreads();

    // ---- GEMM1 (WMMA f32): agg1[m,d] = (sum_n adj[n,m]*bz_s[n,d]) * ndst[m] ---
    {
        const int m0 = wave * 16;
        v8f acc0 = {}, acc1 = {}, acc2 = {}, acc3 = {};
        for (int k0 = 0; k0 < GN; k0 += 4) {
            const int k2 = (k0 >> 1) + hi;            // K-pair row for this half-wave
            v2f a  = *(const v2f*)&adjp[(size_t)(k2 * GN + m0 + l16) * 2];
            const float* brow = &s_bz[k2 * BZ_STRIDE + l16 * 2];
            v2f b0 = *(const v2f*)&brow[0];
            v2f b1 = *(const v2f*)&brow[32];
            v2f b2 = *(const v2f*)&brow[64];
            v2f b3 = *(const v2f*)&brow[96];
            acc0 = __builtin_amdgcn_wmma_f32_16x16x4_f32(false, a, false, b0, (short)0, acc0, false, false);
            acc1 = __builtin_amdgcn_wmma_f32_16x16x4_f32(false, a, false, b1, (short)0, acc1, false, false);
            acc2 = __builtin_amdgcn_wmma_f32_16x16x4_f32(false, a, false, b2, (short)0, acc2, false, false);
            acc3 = __builtin_amdgcn_wmma_f32_16x16x4_f32(false, a, false, b3, (short)0, acc3, false, false);
        }
#pragma unroll
        for (int r = 0; r < 8; ++r) {                 // C/D: VGPR r -> M = r + 8*hi
            int m = m0 + r + hi * 8;
            float sc = ndst[m];
            s_buf[m * AG_STRIDE +  0 + l16] = acc0[r] * sc;
            s_buf[m * AG_STRIDE + 16 + l16] = acc1[r] * sc;
            s_buf[m * AG_STRIDE + 32 + l16] = acc2[r] * sc;
            s_buf[m * AG_STRIDE + 48 + l16] = acc3[r] * sc;
        }
    }
    __syncthreads();

    // ---- GEMM2 (WMMA f32): h1s = tanh(agg1 @ gc1_W + b) * nsrc ---------------
    float* s_h1 = s_bz;   // reuse bz region (pair-packed over m, stride H1_STRIDE)
    {
        const int m0 = wave * 16;
        v8f acc0 = {}, acc1 = {};
        for (int k0 = 0; k0 < GD; k0 += 4) {
            const int ka = k0 + kh;                   // even -> aligned b64
            v2f a  = *(const v2f*)&s_buf[(m0 + l16) * AG_STRIDE + ka];
            const int k2 = (k0 >> 1) + hi;
            v2f b0 = *(const v2f*)&s_w1p[k2 * 64 + l16 * 2];
            v2f b1 = *(const v2f*)&s_w1p[k2 * 64 + 32 + l16 * 2];
            acc0 = __builtin_amdgcn_wmma_f32_16x16x4_f32(false, a, false, b0, (short)0, acc0, false, false);
            acc1 = __builtin_amdgcn_wmma_f32_16x16x4_f32(false, a, false, b1, (short)0, acc1, false, false);
        }
        float bj0 = gc1_b[l16], bj1 = gc1_b[16 + l16];
#pragma unroll
        for (int r = 0; r < 8; ++r) {
            int m = m0 + r + hi * 8;
            float ns = nsrc[m];
            s_h1[(m >> 1) * H1_STRIDE + (0  + l16) * 2 + (m & 1)] = tanhf(acc0[r] + bj0) * ns;
            s_h1[(m >> 1) * H1_STRIDE + (16 + l16) * 2 + (m & 1)] = tanhf(acc1[r] + bj1) * ns;
        }
    }
    __syncthreads();

    // ---- GEMM3 (WMMA f32): agg2[m,j] = (sum_n adj[n,m]*h1s[n,j]) * ndst[m] ---
    float* s_agg2 = s_buf;  // reuse (stride 33)
    {
        const int m0 = wave * 16;
        v8f acc0 = {}, acc1 = {};
        for (int k0 = 0; k0 < GN; k0 += 4) {
            const int k2 = (k0 >> 1) + hi;
            v2f a  = *(const v2f*)&adjp[(size_t)(k2 * GN + m0 + l16) * 2];
            v2f b0 = *(const v2f*)&s_h1[k2 * H1_STRIDE + l16 * 2];
            v2f b1 = *(const v2f*)&s_h1[k2 * H1_STRIDE + 32 + l16 * 2];
            acc0 = __builtin_amdgcn_wmma_f32_16x16x4_f32(false, a, false, b0, (short)0, acc0, false, false);
            acc1 = __builtin_amdgcn_wmma_f32_16x16x4_f32(false, a, false, b1, (short)0, acc1, false, false);
        }
#pragma unroll
        for (int r = 0; r < 8; ++r) {
            int m = m0 + r + hi * 8;
            float sc = ndst[m];
            s_agg2[m * 33 +  0 + l16] = acc0[r] * sc;
            s_agg2[m * 33 + 16 + l16] = acc1[r] * sc;
        }
    }
    __syncthreads();

    // ---- final 32-wide projection: x_hat = agg2 @ gc2_W + gc2_b --------------
    if (t < GN) {
        float o = gc2_b[0];
        for (int j = 0; j < 32; ++j) o = fmaf(s_agg2[t * 33 + j], gc2_W[j], o);
        out[b * GN + t] = o;
    }
}

// ---------------------------------------------------------------------------
extern "C" void kernel_launch(void* const* d_in, const int* in_sizes, int n_in,
                              void* d_out, int out_size, void* d_ws, size_t ws_size,
                              hipStream_t stream) {
    const float* x      = (const float*)d_in[0];
    const float* eps    = (const float*)d_in[1];
    const float* gl     = (const float*)d_in[2];
    const float* W_enc  = (const float*)d_in[3];
    const float* b_enc  = (const float*)d_in[4];
    const float* W_mu   = (const float*)d_in[5];
    const float* b_mu   = (const float*)d_in[6];
    const float* W_lv   = (const float*)d_in[7];
    const float* b_lv   = (const float*)d_in[8];
    const float* emb_W  = (const float*)d_in[9];
    const float* emb_b  = (const float*)d_in[10];
    const float* gc1_W  = (const float*)d_in[11];
    const float* gc1_b  = (const float*)d_in[12];
    const float* gc2_W  = (const float*)d_in[13];
    const float* gc2_b  = (const float*)d_in[14];
    const int*   it_num = (const int*)d_in[15];

    float* ws   = (float*)d_ws;
    float* adj  = ws + WS_ADJ;
    float* adjp = ws + WS_ADJP;
    float* ns   = ws + WS_NSRC;
    float* nd   = ws + WS_NDST;
    float* W0   = ws + WS_W0;
    float* Cm   = ws + WS_CMAT;
    float* out  = (float*)d_out;

    gg_build_adj  <<<GN, GN, 0, stream>>>(gl, it_num, adj, adjp);
    gg_build_norms<<<1,  GN, 0, stream>>>(adj, ns, nd);
    gg_build_emb  <<<64, GN, 0, stream>>>(emb_W, emb_b, W0, Cm);
    gg_main<<<1024, 512, 0, stream>>>(x, eps, W_enc, b_enc, W_mu, b_mu, W_lv, b_lv,
                                      gc1_W, gc1_b, gc2_W, gc2_b,
                                      adjp, ns, nd, W0, Cm, out);
}